// DFFN_26766236188755
// MI455X (gfx1250) — compile-verified
//
#include <hip/hip_runtime.h>
#include <math.h>
#include <stdint.h>

// ---------------------------------------------------------------------------
// DFFN on gfx1250 (MI455X): fp32 WMMA (16x16x4) for all three GEMM stages.
//   spectral gating  ==  per-channel circular conv == GEMM with circulant H_c
// Slab pipeline (8 slabs of 32 rows, 8-row patch halo) keeps workspace ~82MB.
// All WMMA operand pairs are stored contiguously -> single b64 loads, no movs.
// Bulk tile staging uses gfx1250 async global<->LDS copies (ASYNCcnt).
// ---------------------------------------------------------------------------

typedef float v2f __attribute__((ext_vector_type(2)));
typedef float v8f __attribute__((ext_vector_type(8)));

#define WMMA_F32(a, b, c) \
    __builtin_amdgcn_wmma_f32_16x16x4_f32(false, (a), false, (b), (short)0, (c), false, false)

// async copy helpers: GVS mode, mem = SADDR(64) + VADDR(32) + IOFFSET
__device__ __forceinline__ void async_g2l_b32(unsigned lds_byte, unsigned goff_byte,
                                              const void* base) {
    asm volatile("global_load_async_to_lds_b32 %0, %1, %2"
                 :: "v"(lds_byte), "v"(goff_byte), "s"(base) : "memory");
}
__device__ __forceinline__ void async_l2g_b32(unsigned goff_byte, unsigned lds_byte,
                                              void* base) {
    asm volatile("global_store_async_from_lds_b32 %0, %1, %2"
                 :: "v"(goff_byte), "v"(lds_byte), "s"(base) : "memory");
}
__device__ __forceinline__ void wait_async0() {
    asm volatile("s_wait_asynccnt 0x0" ::: "memory");
}
__device__ __forceinline__ void wait_ds0() {
    asm volatile("s_wait_dscnt 0x0" ::: "memory");
}

// workspace layout (floats):
//   HT   : 384 * 64 * 64   paired circulants HT[c][m>>1][n][m&1]; B[m][n]=h[(n-m) mod 8/dim]
//   gbuf : 4 * 384 * 48 * 256   slab buffer rows [r0-8, r0+40)
#define HT_FLOATS   (384 * 4096)
#define GB_IDX(b, c, gr, w) ((((b) * 384 + (c)) * 48 + (gr)) * 256 + (w))

// ---------------------------------------------------------------------------
// K0: build per-channel circulant matrix from rfft filter (8x5).
// irfft2(rfft2(x) * F) == x (circ-conv) h, h = IDFT2(M), M = hermitian ext of F
// (DC / Nyquist columns symmetrized in k1 -> pocketfft c2r semantics).
// All twiddles are cos(2*pi*j/8): exact 8-entry table.
// Output in K-pair-contiguous layout so K2 B-frags are single b64 loads.
// ---------------------------------------------------------------------------
__global__ void dffn_build_ht(const float* __restrict__ filt, float* __restrict__ HT) {
    int c = blockIdx.x;   // 384
    int t = threadIdx.x;  // 64
    __shared__ float hsp[64];
    const float ct[8] = {1.f, 0.70710678118654752f, 0.f, -0.70710678118654752f,
                         -1.f, -0.70710678118654752f, 0.f, 0.70710678118654752f};
    const float* F = filt + c * 40;  // (8,5) row-major
    int d1 = t >> 3, d2 = t & 7;
    float acc = 0.f;
#pragma unroll
    for (int k1 = 0; k1 < 8; ++k1) {
#pragma unroll
        for (int k2 = 0; k2 < 8; ++k2) {
            int mk1 = (8 - k1) & 7;
            float m;
            if (k2 >= 1 && k2 <= 3)      m = F[k1 * 5 + k2];
            else if (k2 >= 5)            m = F[mk1 * 5 + (8 - k2)];
            else /* k2 == 0 or 4 */      m = 0.5f * (F[k1 * 5 + k2] + F[mk1 * 5 + k2]);
            acc += m * ct[(k1 * d1 + k2 * d2) & 7];
        }
    }
    hsp[t] = acc * (1.f / 64.f);
    __syncthreads();
    for (int i = t; i < 4096; i += 64) {
        int m = i >> 6, n = i & 63;
        int dd1 = ((n >> 3) - (m >> 3)) & 7;
        int dd2 = ((n & 7) - (m & 7)) & 7;
        // paired: HT[c][m>>1][n][m&1]
        HT[c * 4096 + (((m >> 1) * 64 + n) << 1) + (m & 1)] = hsp[dd1 * 8 + dd2];
    }
}

// ---------------------------------------------------------------------------
// K1: project_in GEMM.  Per block: M=128 pixels (half image row) x N=192 x K=64.
// dyn LDS: A 128*66=8448 | B paired 12288 | stage 2048 = 22784 floats (91136 B)
// A/B tiles staged with async global->LDS copies.
// ---------------------------------------------------------------------------
__global__ __launch_bounds__(256) void dffn_proj_in(const float* __restrict__ x,
                                                    const float* __restrict__ w_in,
                                                    float* __restrict__ gbuf, int r0) {
    extern __shared__ float sm[];
    float* Atile = sm;                 // [128 px][66 pad] : c-pairs contiguous
    float* Btile = sm + 8448;          // [32 cpair][192 o][2]
    float* stg   = sm + 8448 + 12288;  // [16 o][128 px]

    int bid = blockIdx.x;              // 768 = 2(N) * 2(col) * 48(row) * 4(b)
    int nhalf = bid & 1;
    int chalf = (bid >> 1) & 1;
    int rowi  = (bid >> 2) % 48;
    int b     = bid / 192;
    int row = r0 - 8 + rowi;
    if (row < 0 || row >= 256) return;  // uniform per block
    int col0  = chalf * 128;
    int obase = nhalf * 192;
    int t = threadIdx.x;
    unsigned lbase = __builtin_amdgcn_groupstaticsize();

    for (int i = t; i < 8192; i += 256) {
        int c = i >> 7, px = i & 127;
        unsigned lds  = lbase + 4u * (unsigned)(px * 66 + c);
        unsigned goff = 4u * (unsigned)(((b * 64 + c) * 256 + row) * 256 + col0 + px);
        async_g2l_b32(lds, goff, x);
    }
    for (int i = t; i < 12288; i += 256) {
        int o = i >> 6, ci = i & 63;
        unsigned lds  = lbase + 4u * (unsigned)(8448 + (((ci >> 1) * 192 + o) << 1) + (ci & 1));
        unsigned goff = 4u * (unsigned)((obase + o) * 64 + ci);
        async_g2l_b32(lds, goff, w_in);
    }
    wait_async0();
    __syncthreads();

    int wave = t >> 5, lane = t & 31;
    int lm = lane & 15;
    int koff = (lane >> 4) * 2;
    v8f vzero = {0.f, 0.f, 0.f, 0.f, 0.f, 0.f, 0.f, 0.f};
    v8f acc[12];
#pragma unroll
    for (int i = 0; i < 12; ++i) acc[i] = vzero;
    int mpx = wave * 16 + lm;  // wave w owns M-tile w

    for (int kk = 0; kk < 16; ++kk) {
        int c0 = kk * 4 + koff;
        v2f a = *(const v2f*)&Atile[mpx * 66 + c0];
#pragma unroll
        for (int nt = 0; nt < 12; ++nt) {
            int o = nt * 16 + lm;
            v2f bb = *(const v2f*)&Btile[(((c0 >> 1) * 192 + o) << 1)];
            acc[nt] = WMMA_F32(a, bb, acc[nt]);
        }
    }

    // restage each 128x16 N-tile through LDS for coalesced global stores
    for (int nt = 0; nt < 12; ++nt) {
        __syncthreads();
#pragma unroll
        for (int i = 0; i < 8; ++i) {
            int m = i + ((lane < 16) ? 0 : 8);
            stg[lm * 128 + wave * 16 + m] = acc[nt][i];
        }
        __syncthreads();
#pragma unroll
        for (int j = 0; j < 8; ++j) {
            int flat = t * 8 + j;
            int ol = flat >> 7, px = flat & 127;
            gbuf[GB_IDX(b, obase + nt * 16 + ol, rowi, col0 + px)] = stg[ol * 128 + px];
        }
    }
}

// ---------------------------------------------------------------------------
// K2: spectral gating, in-place on gbuf. One (channel, 16-patch strip) / wave.
// GEMM: M=16 patches x N=64 pix x K=64 pix, B = paired circulant (global b64,
// L2 resident). In/out tile moved with async global<->LDS copies.
// dyn LDS per wave: A(16x68)=1088 + D(16x64)=1024 -> 8*2112 floats (67584 B).
// No block barriers (waves may early-out on image edges).
// ---------------------------------------------------------------------------
__global__ __launch_bounds__(256) void dffn_gate(float* __restrict__ gbuf,
                                                 const float* __restrict__ HT, int r0) {
    extern __shared__ float sm[];
    int t = threadIdx.x;
    int wave = t >> 5, lane = t & 31;
    float* Aw = sm + wave * 2112;
    unsigned lbase = __builtin_amdgcn_groupstaticsize() + 4u * (unsigned)(wave * 2112);

    int tid = blockIdx.x * 8 + wave;      // 2304 blocks -> 18432 wave tasks
    int strip = tid % 12;                 // 6 patch rows * 2 halves
    int c = (tid / 12) % 384;
    int b = tid / (12 * 384);
    int prow = strip >> 1, half = strip & 1;
    int gpr = (r0 >> 3) - 1 + prow;       // global patch row
    if (gpr < 0 || gpr >= 32) return;     // wave-level, no barriers in kernel

    int rowbase = GB_IDX(b, c, prow * 8, half * 128);

    // async load: 16 patches x 64 pix, permuted to [patch][pix] (stride 68)
#pragma unroll
    for (int j = 0; j < 32; ++j) {
        int flat = j * 32 + lane;
        int rr = flat >> 7, xx = flat & 127;
        unsigned lds  = lbase + 4u * (unsigned)((xx >> 3) * 68 + rr * 8 + (xx & 7));
        unsigned goff = 4u * (unsigned)(rowbase + rr * 256 + xx);
        async_g2l_b32(lds, goff, gbuf);
    }
    wait_async0();

    v8f vzero = {0.f, 0.f, 0.f, 0.f, 0.f, 0.f, 0.f, 0.f};
    v8f acc[4];
#pragma unroll
    for (int i = 0; i < 4; ++i) acc[i] = vzero;
    int lm = lane & 15, koff = (lane >> 4) * 2;
    const v2f* HTp = (const v2f*)HT + c * 2048;  // [kpair][n]

    for (int kk = 0; kk < 16; ++kk) {
        int q0 = kk * 4 + koff;
        v2f a = *(const v2f*)&Aw[lm * 68 + q0];
#pragma unroll
        for (int nt = 0; nt < 4; ++nt) {
            int n = nt * 16 + lm;
            v2f bb = HTp[(q0 >> 1) * 64 + n];
            acc[nt] = WMMA_F32(a, bb, acc[nt]);
        }
    }

    float* Dw = Aw + 1088;
#pragma unroll
    for (int nt = 0; nt < 4; ++nt) {
#pragma unroll
        for (int i = 0; i < 8; ++i) {
            int m = i + ((lane < 16) ? 0 : 8);
            Dw[m * 64 + nt * 16 + lm] = acc[nt][i];
        }
    }
    wait_ds0();  // D tile resident in LDS before async engine reads it
#pragma unroll
    for (int j = 0; j < 32; ++j) {
        int flat = j * 32 + lane;
        int rr = flat >> 7, xx = flat & 127;
        unsigned lds  = lbase + 4u * (unsigned)(1088 + (xx >> 3) * 64 + rr * 8 + (xx & 7));
        unsigned goff = 4u * (unsigned)(rowbase + rr * 256 + xx);
        async_l2g_b32(goff, lds, gbuf);
    }
    // s_endpgm performs an implicit wait-idle (covers ASYNCcnt)
}

// ---------------------------------------------------------------------------
// K3: depthwise 3x3 + exact GELU gate + project_out GEMM. One 8x8 patch / block.
// dyn LDS: halo 192*100=19200 | z 64*196=12544 | wdw 3456  (140800 B);
// output restaged into the halo region after it is dead.
// B-frags come straight from w_out as contiguous float2 (global b64, L2 hot).
// ---------------------------------------------------------------------------
__global__ __launch_bounds__(256) void dffn_back(const float* __restrict__ gbuf,
                                                 const float* __restrict__ w_dw,
                                                 const float* __restrict__ w_out,
                                                 float* __restrict__ out, int r0) {
    extern __shared__ float sm[];
    float* halo = sm;                   // [192 ch][10*10]
    float* z    = sm + 19200;           // [64 px][196 pad] (192 used)
    float* wdw  = sm + 19200 + 12544;   // 384*9

    int bid = blockIdx.x;               // 512 = 32 pc * 4 pr * 4 b
    int pc = bid & 31, pri = (bid >> 5) & 3, b = bid >> 7;
    int pr = (r0 >> 3) + pri;
    int h0 = pr * 8, w0 = pc * 8;
    int t = threadIdx.x;

    for (int i = t; i < 3456; i += 256) wdw[i] = w_dw[i];

    for (int cb = 0; cb < 2; ++cb) {    // channel pair-blocks: {cb*96..+95, 192+cb*96..+95}
        __syncthreads();
        for (int i = t; i < 19200; i += 256) {
            int hc = i / 100, q = i - hc * 100;
            int hy = q / 10, hx = q - hy * 10;
            int c = (hc < 96) ? (cb * 96 + hc) : (192 + cb * 96 + (hc - 96));
            int h = h0 - 1 + hy, w = w0 - 1 + hx;
            float v = 0.f;
            if (h >= 0 && h < 256 && w >= 0 && w < 256)
                v = gbuf[GB_IDX(b, c, h - r0 + 8, w)];
            halo[i] = v;
        }
        __syncthreads();
        for (int i = t; i < 6144; i += 256) {
            int px = i & 63, j = i >> 6;   // j in [0,96)
            int r = px >> 3, cc = px & 7;
            float v1 = 0.f, v2 = 0.f;
#pragma unroll
            for (int ky = 0; ky < 3; ++ky) {
#pragma unroll
                for (int kx = 0; kx < 3; ++kx) {
                    float hv1 = halo[j * 100 + (r + ky) * 10 + (cc + kx)];
                    float hv2 = halo[(96 + j) * 100 + (r + ky) * 10 + (cc + kx)];
                    v1 += wdw[(cb * 96 + j) * 9 + ky * 3 + kx] * hv1;
                    v2 += wdw[(192 + cb * 96 + j) * 9 + ky * 3 + kx] * hv2;
                }
            }
            float g = 0.5f * v1 * (1.f + erff(v1 * 0.70710678118654752f));  // exact GELU
            z[px * 196 + (cb * 96 + j)] = g * v2;
        }
    }
    __syncthreads();

    // GEMM: M=64 px x N=64 o x K=192; wave w -> (mtile = w>>1, ntiles = (w&1)*2 + {0,1})
    int wave = t >> 5, lane = t & 31;
    int lm = lane & 15, koff = (lane >> 4) * 2;
    int mt = wave >> 1;
    int ntb = (wave & 1) * 2;
    v8f vzero = {0.f, 0.f, 0.f, 0.f, 0.f, 0.f, 0.f, 0.f};
    v8f acc[2] = {vzero, vzero};

    for (int kk = 0; kk < 48; ++kk) {
        int c0 = kk * 4 + koff;
        int prix = mt * 16 + lm;
        v2f a = *(const v2f*)&z[prix * 196 + c0];
#pragma unroll
        for (int u = 0; u < 2; ++u) {
            int o = (ntb + u) * 16 + lm;
            v2f bb = *(const v2f*)&w_out[o * 192 + c0];  // contiguous pair, L2 hot
            acc[u] = WMMA_F32(a, bb, acc[u]);
        }
    }

    float* ost = sm;  // reuse dead halo region: [64 o][64 px]
#pragma unroll
    for (int u = 0; u < 2; ++u) {
#pragma unroll
        for (int i = 0; i < 8; ++i) {
            int m = i + ((lane < 16) ? 0 : 8);
            ost[((ntb + u) * 16 + lm) * 64 + mt * 16 + m] = acc[u][i];
        }
    }
    __syncthreads();
    for (int i = t; i < 4096; i += 256) {
        int o = i >> 6, px = i & 63;
        int r = px >> 3, cc = px & 7;
        out[((b * 64 + o) * 256 + h0 + r) * 256 + w0 + cc] = ost[i];
    }
}

// ---------------------------------------------------------------------------
extern "C" void kernel_launch(void* const* d_in, const int* in_sizes, int n_in,
                              void* d_out, int out_size, void* d_ws, size_t ws_size,
                              hipStream_t stream) {
    (void)in_sizes; (void)n_in; (void)out_size; (void)ws_size;
    const float* x     = (const float*)d_in[0];
    const float* w_in  = (const float*)d_in[1];
    const float* w_dw  = (const float*)d_in[2];
    const float* filt  = (const float*)d_in[3];
    const float* w_out = (const float*)d_in[4];
    float* out  = (float*)d_out;
    float* HT   = (float*)d_ws;
    float* gbuf = HT + HT_FLOATS;   // needs ~82 MB workspace total

    dffn_build_ht<<<384, 64, 0, stream>>>(filt, HT);
    for (int s = 0; s < 8; ++s) {
        int r0 = s * 32;
        dffn_proj_in<<<768, 256, 22784 * 4, stream>>>(x, w_in, gbuf, r0);
        dffn_gate<<<2304, 256, 16896 * 4, stream>>>(gbuf, HT, r0);
        dffn_back<<<512, 256, 35200 * 4, stream>>>(gbuf, w_dw, w_out, out, r0);
    }
}